// DETRPostProcessor_44796508897684
// MI455X (gfx1250) — compile-verified
//
#include <hip/hip_runtime.h>
#include <math.h>

// DETR post-process for MI455X (gfx1250), wave32.
// One 512-thread WG (16 waves) per batch row. Double-buffered async
// global->LDS streaming (ASYNCcnt-tracked, no per-tile block barriers since
// each thread reads only its own staged slot); 12-bit radix-select threshold;
// bitonic sort of <=2048 candidates; exact sigmoid on the 300 survivors.

#define NQ    1000
#define NK    80
#define ROW   (NQ * NK)      // 80000 logits per batch row
#define ROW4  (ROW / 4)      // 20000 float4
#define TOPK  300
#define NBINS 4096
#define CAP   2048
#define BDIM  512
#define NTILES ((ROW4 + BDIM - 1) / BDIM)   // 40

struct Smem {
  unsigned           hist[NBINS];      // 16 KB
  unsigned           scan[BDIM];       //  2 KB
  unsigned long long cand[CAP];        // 16 KB
  float4             stage[2][BDIM];   // 16 KB double-buffered staging
  unsigned           cnt;
  unsigned           thrBin;
};

// Monotone map: descending float order == descending unsigned order.
__device__ __forceinline__ unsigned order_key(float f) {
  unsigned u = __float_as_uint(f);
  return (u & 0x80000000u) ? ~u : (u | 0x80000000u);
}

// LDS byte offset of a __shared__ object (flat shared addr low 32 bits).
__device__ __forceinline__ unsigned lds_off_u32(const void* p) {
  return (unsigned)(unsigned long long)p;
}

// Per-lane 16B async copy: global address base plus voff to the LDS address in lds_addr (GVS addressing)
__device__ __forceinline__ void async_copy_b128(unsigned lds_addr, unsigned voff,
                                                const float* base) {
  asm volatile("global_load_async_to_lds_b128 %0, %1, %2"
               :: "v"(lds_addr), "v"(voff), "s"(base)
               : "memory");
}
__device__ __forceinline__ void wait_async_le1() {
  asm volatile("s_wait_asynccnt 0x1" ::: "memory");
}
__device__ __forceinline__ void wait_async_le0() {
  asm volatile("s_wait_asynccnt 0x0" ::: "memory");
}

__global__ __launch_bounds__(BDIM)
void detr_post_kernel(const float* __restrict__ logits,
                      const float* __restrict__ boxes,
                      const int*   __restrict__ osz,
                      float*       __restrict__ out) {
  __shared__ Smem sm;
  const int n   = blockIdx.x;
  const int tid = threadIdx.x;
  const float* lrow = logits + (size_t)n * ROW;

  for (int b = tid; b < NBINS; b += BDIM) sm.hist[b] = 0u;
  if (tid == 0) { sm.cnt = 0u; sm.thrBin = 0u; }

  const unsigned lds0 = lds_off_u32(&sm.stage[0][0]) + (unsigned)tid * 16u;
  const unsigned lds1 = lds_off_u32(&sm.stage[1][0]) + (unsigned)tid * 16u;
  __syncthreads();   // hist zeroed before any atomics

  // ---- Pass 1: histogram of top-12 key bits; double-buffered async stream.
  // Each thread reads back only its own staged slot, so only the per-wave
  // ASYNCcnt wait orders LDS reads -- no per-tile block barriers needed.
  if (tid < ROW4) async_copy_b128(lds0, (unsigned)tid * 16u, lrow);
  for (int t = 0; t < NTILES; ++t) {
    const int g4n = (t + 1) * BDIM + tid;
    if (t + 1 < NTILES) {
      if (g4n < ROW4)
        async_copy_b128(((t + 1) & 1) ? lds1 : lds0, (unsigned)g4n * 16u, lrow);
      wait_async_le1();
    } else {
      wait_async_le0();
    }
    const int g4 = t * BDIM + tid;
    if (g4 < ROW4) {
      float4 v = sm.stage[t & 1][tid];
      atomicAdd(&sm.hist[order_key(v.x) >> 20], 1u);
      atomicAdd(&sm.hist[order_key(v.y) >> 20], 1u);
      atomicAdd(&sm.hist[order_key(v.z) >> 20], 1u);
      atomicAdd(&sm.hist[order_key(v.w) >> 20], 1u);
    }
  }
  __syncthreads();   // histogram complete across all waves

  // Prefetch pass-2 tile 0 now; threshold scan below overlaps the fetch.
  if (tid < ROW4) async_copy_b128(lds0, (unsigned)tid * 16u, lrow);

  // ---- Threshold: descending cumulative count crosses TOPK ----
  const int hi = NBINS - 1 - tid * 8;        // this thread's 8 bins, descending
  unsigned psum = 0;
  #pragma unroll
  for (int j = 0; j < 8; ++j) psum += sm.hist[hi - j];
  sm.scan[tid] = psum;
  __syncthreads();
  for (int s = 1; s < BDIM; s <<= 1) {       // Hillis-Steele inclusive scan
    unsigned v = (tid >= s) ? sm.scan[tid - s] : 0u;
    __syncthreads();
    sm.scan[tid] += v;
    __syncthreads();
  }
  const unsigned incl = sm.scan[tid];
  const unsigned excl = incl - psum;
  if (excl < TOPK && incl >= TOPK) {         // exactly one thread matches
    unsigned run = excl;
    #pragma unroll
    for (int j = 0; j < 8; ++j) {
      unsigned c = sm.hist[hi - j];
      if (run + c >= TOPK) { sm.thrBin = (unsigned)(hi - j); break; }
      run += c;
    }
  }
  __syncthreads();
  const unsigned T = sm.thrBin;

  // ---- Pass 2: collect candidates with bin >= T (L2-resident restream) ----
  for (int t = 0; t < NTILES; ++t) {
    const int g4n = (t + 1) * BDIM + tid;
    if (t + 1 < NTILES) {
      if (g4n < ROW4)
        async_copy_b128(((t + 1) & 1) ? lds1 : lds0, (unsigned)g4n * 16u, lrow);
      wait_async_le1();
    } else {
      wait_async_le0();
    }
    const int g4 = t * BDIM + tid;
    if (g4 < ROW4) {
      float4 v = sm.stage[t & 1][tid];
      float vals[4] = {v.x, v.y, v.z, v.w};
      #pragma unroll
      for (int e = 0; e < 4; ++e) {
        unsigned u = order_key(vals[e]);
        if ((u >> 20) >= T) {
          unsigned idx  = (unsigned)(g4 * 4 + e);
          unsigned slot = atomicAdd(&sm.cnt, 1u);
          if (slot < CAP)  // key: value desc, ties -> smaller index first
            sm.cand[slot] = ((unsigned long long)u << 32) | (unsigned)(~idx);
        }
      }
    }
  }
  __syncthreads();   // all appends visible
  unsigned cnt = sm.cnt;
  if (cnt > CAP) cnt = CAP;

  // ---- Bitonic sort descending over next-pow2 padded candidates ----
  unsigned m = 2;
  while (m < cnt) m <<= 1;
  for (unsigned i = cnt + tid; i < m; i += BDIM) sm.cand[i] = 0ull;
  __syncthreads();
  for (unsigned k = 2; k <= m; k <<= 1) {
    for (unsigned j = k >> 1; j > 0; j >>= 1) {
      for (unsigned i = tid; i < m; i += BDIM) {
        unsigned ixj = i ^ j;
        if (ixj > i) {
          unsigned long long a = sm.cand[i], b = sm.cand[ixj];
          bool desc = ((i & k) == 0);
          if (desc ? (a < b) : (a > b)) { sm.cand[i] = b; sm.cand[ixj] = a; }
        }
      }
      __syncthreads();
    }
  }

  // ---- Emit top-300: [label, sigmoid(score), x, y, w, h] ----
  const float sy = (float)osz[0];   // original_sizes[0] = (h, w) for ALL rows
  const float sx = (float)osz[1];   // scale = [w, h, w, h]
  for (int i = tid; i < TOPK; i += BDIM) {
    unsigned long long key = sm.cand[i];
    unsigned idx = ~(unsigned)(key & 0xFFFFFFFFull);
    unsigned lab = idx % NK;
    unsigned q   = idx / NK;
    float logit  = lrow[idx];
    float score  = 1.0f / (1.0f + expf(-logit));
    const float* bp = boxes + ((size_t)n * NQ + q) * 4;
    float cx = bp[0], cy = bp[1], w = bp[2], h = bp[3];
    float* op = out + ((size_t)n * TOPK + (size_t)i) * 6;
    op[0] = (float)lab;
    op[1] = score;
    op[2] = (cx - 0.5f * w) * sx;
    op[3] = (cy - 0.5f * h) * sy;
    op[4] = w * sx;
    op[5] = h * sy;
  }
}

extern "C" void kernel_launch(void* const* d_in, const int* in_sizes, int n_in,
                              void* d_out, int out_size, void* d_ws, size_t ws_size,
                              hipStream_t stream) {
  const float* logits = (const float*)d_in[0];   // (N, 1000, 80) f32
  const float* boxes  = (const float*)d_in[1];   // (N, 1000, 4)  f32
  const int*   osz    = (const int*)d_in[2];     // (N, 2) i32
  float* out = (float*)d_out;                    // (N, 300, 6) f32
  const int N = in_sizes[0] / ROW;               // 256
  (void)n_in; (void)out_size; (void)d_ws; (void)ws_size;
  detr_post_kernel<<<dim3(N), dim3(BDIM), 0, stream>>>(logits, boxes, osz, out);
}